// FullyConnected_91053306675325
// MI455X (gfx1250) — compile-verified
//
#include <hip/hip_runtime.h>
#include <hip/hip_bf16.h>
#include <math.h>

typedef __attribute__((ext_vector_type(2))) float v2f;
typedef __attribute__((ext_vector_type(8))) float v8f;

#define DD    46
#define KC    92          // 2*DD
#define H1    256
#define H2C   128
#define NI    4000
#define NJ    100
#define NPIX  400000      // NI*NJ
#define NTILE 16
#define NT_TOTAL 25000    // NPIX/16
#define NTPB  10
#define NBLK  2500        // NT_TOTAL/NTPB
#define NSEL  50

// workspace layout (float offsets)
constexpr int OF_S0    = 0;        // 46
constexpr int OF_MEANH = 64;       // 46
constexpr int OF_Z1HF  = 128;      // 46*50
constexpr int OF_Z1LF  = 2560;     // 46*50
constexpr int OF_A1    = 5120;     // 256
constexpr int OF_D1    = 5376;     // 256
constexpr int OF_B2F   = 5632;     // 128
constexpr int OF_A2    = 5888;     // 128
constexpr int OF_D2    = 6144;     // 128
constexpr int OF_W2F   = 8192;     // 128*256
constexpr int OF_Z0S   = 49152;    // 46*4000
constexpr int OF_Z0MD  = 233216;   // 46*4000
constexpr int OF_Z0MM  = 417280;   // 46*4000
constexpr int OF_P1S   = 601344;   // NBLK*256
constexpr int OF_P1Q   = 1241344;  // NBLK*256
constexpr int OF_P2S   = 1881344;  // NBLK*128
constexpr int OF_P2Q   = 2201344;  // NBLK*128

__device__ __forceinline__ float sigm(float x) { return 1.0f / (1.0f + expf(-x)); }

__device__ __forceinline__ float conv3(const float* y, const float* w, int c) {
    float l = (c > 0)      ? y[c - 1] : 0.0f;
    float r = (c < DD - 1) ? y[c + 1] : 0.0f;
    return w[0] * l + w[1] * y[c] + w[2] * r;
}

__device__ __forceinline__ v8f wmma4(v2f a, v2f b, v8f c) {
    return __builtin_amdgcn_wmma_f32_16x16x4_f32(false, a, false, b, (short)0, c, false, false);
}

// ---------------- K1: ECA scales + pairwise-diff reductions (tiny) ----------------
__global__ __launch_bounds__(512)
void k_pre1(const float* __restrict__ z0, const float* __restrict__ z1,
            const float* __restrict__ we1, const float* __restrict__ we2,
            const float* __restrict__ we3, const float* __restrict__ we4,
            const float* __restrict__ we5,
            const int* __restrict__ cdrH, const int* __restrict__ cdrL,
            const int* __restrict__ notH, const int* __restrict__ notL,
            int nHc, int nHn, int nLc, int nLn,
            float* __restrict__ ws)
{
    __shared__ float y0[DD], yH[DD], yL[DD];
    __shared__ float s0[DD], sH46[DD], sL46[DD], s4[DD], s5[DD];
    __shared__ float MH[DD * NSEL], ML[DD * NSEL];
    __shared__ float red[512];
    int t = threadIdx.x;
    int c8 = t >> 3, sub = t & 7;

    float p0 = 0.f, pH = 0.f, pL = 0.f;
    if (c8 < DD) {
        for (int i = sub; i < NI; i += 8)       p0 += z0[i * DD + c8];
        for (int i = sub; i < 250; i += 8)      pH += z1[i * DD + c8];
        for (int i = 250 + sub; i < 600; i += 8) pL += z1[i * DD + c8];
    }
    red[t] = p0; __syncthreads();
    if (t < DD) { float s = 0.f; for (int k = 0; k < 8; ++k) s += red[t * 8 + k]; y0[t] = s / (float)NI; }
    __syncthreads();
    red[t] = pH; __syncthreads();
    if (t < DD) { float s = 0.f; for (int k = 0; k < 8; ++k) s += red[t * 8 + k]; yH[t] = s / 250.f; }
    __syncthreads();
    red[t] = pL; __syncthreads();
    if (t < DD) { float s = 0.f; for (int k = 0; k < 8; ++k) s += red[t * 8 + k]; yL[t] = s / 350.f; }
    __syncthreads();
    if (t < DD) {
        s0[t]   = sigm(conv3(y0, we1, t));
        sH46[t] = sigm(conv3(yH, we2, t));
        sL46[t] = sigm(conv3(yL, we3, t));
    }
    __syncthreads();
    // M_H[c][a] = sH[c] * mean_b | z1H[c,cdrH[a]] - z1H[c,notH[b]] |
    for (int idx = t; idx < DD * nHc; idx += 512) {
        int c = idx / nHc, a = idx - c * nHc;
        float va = z1[cdrH[a] * DD + c];
        float s = 0.f;
        for (int b = 0; b < nHn; ++b) s += fabsf(va - z1[notH[b] * DD + c]);
        MH[c * NSEL + a] = sH46[c] * (s / (float)nHn);
    }
    for (int idx = t; idx < DD * nLc; idx += 512) {
        int c = idx / nLc, a = idx - c * nLc;
        float va = z1[(250 + cdrL[a]) * DD + c];
        float s = 0.f;
        for (int b = 0; b < nLn; ++b) s += fabsf(va - z1[(250 + notL[b]) * DD + c]);
        ML[c * NSEL + a] = sL46[c] * (s / (float)nLn);
    }
    __syncthreads();
    if (t < DD) {
        float s = 0.f; for (int a = 0; a < nHc; ++a) s += MH[t * NSEL + a]; y0[t] = s / (float)nHc; // y4
        s = 0.f;       for (int a = 0; a < nLc; ++a) s += ML[t * NSEL + a]; yH[t] = s / (float)nLc; // y5
    }
    __syncthreads();
    if (t < DD) {
        s4[t] = sigm(conv3(y0, we4, t));
        s5[t] = sigm(conv3(yH, we5, t));
        ws[OF_S0 + t]    = s0[t];
        ws[OF_MEANH + t] = s4[t] * y0[t];   // mean_a z1Hfinal[c,a]
    }
    __syncthreads();
    for (int idx = t; idx < DD * nHc; idx += 512) {
        int c = idx / nHc, a = idx - c * nHc;
        ws[OF_Z1HF + c * NSEL + a] = s4[c] * MH[c * NSEL + a];
    }
    for (int idx = t; idx < DD * nLc; idx += 512) {
        int c = idx / nLc, a = idx - c * nLc;
        ws[OF_Z1LF + c * NSEL + a] = s5[c] * ML[c * NSEL + a];
    }
}

// ---------------- K2: per-pixel precompute z0s, z0m_dif, z0m_mul ----------------
__global__ __launch_bounds__(64)
void k_pre2(const float* __restrict__ z0, float* __restrict__ ws)
{
    int c = threadIdx.x;
    int i = blockIdx.x;
    if (c >= DD) return;
    float v = z0[i * DD + c] * ws[OF_S0 + c];
    ws[OF_Z0S + c * NI + i] = v;
    float s = 0.f;
    const float* zh = ws + OF_Z1HF + c * NSEL;
    for (int a = 0; a < NSEL; ++a) s += fabsf(v - zh[a]);
    ws[OF_Z0MD + c * NI + i] = s / (float)NSEL;
    ws[OF_Z0MM + c * NI + i] = v * ws[OF_MEANH + c];
}

// ---------------- K3/K5: main fused GEMM kernel ----------------
// Pass A (FULL=false): GEMM1 -> BN1 partial stats only.
// Pass B (FULL=true):  GEMM1 -> LDS, GEMM2 (BN1 folded into W2F/B2F) -> out + BN2 partials.
template<bool FULL>
__global__ __launch_bounds__(256)
void k_main(const float* __restrict__ conv_w, const float* __restrict__ conv_b,
            float* __restrict__ ws, float* __restrict__ out)
{
    __shared__ float zc[KC * NTILE];     // z_cat tile [k][n]
    __shared__ float c1t[H1 * NTILE];    // GEMM1 result tile [ch][n] (pass B)

    const float* z0s  = ws + OF_Z0S;
    const float* z0md = ws + OF_Z0MD;
    const float* z0mm = ws + OF_Z0MM;
    const float* z1hf = ws + OF_Z1HF;
    const float* z1lf = ws + OF_Z1LF;
    const float* w2f  = ws + OF_W2F;
    const float* b2f  = ws + OF_B2F;

    int t    = threadIdx.x;
    int lane = t & 31;
    int wv   = t >> 5;
    int l16  = lane & 15;     // n for B/C/D frags, m for A frags
    int h    = lane >> 4;

    float sA[2][8], qA[2][8];   // pass A: BN1 per-lane partials (2 M-tiles)
    float s2[8],    q2[8];      // pass B: BN2 per-lane partials
#pragma unroll
    for (int u = 0; u < 8; ++u) { sA[0][u] = qA[0][u] = sA[1][u] = qA[1][u] = 0.f; s2[u] = q2[u] = 0.f; }

    for (int tt = 0; tt < NTPB; ++tt) {
        int nt = blockIdx.x * NTPB + tt;
        int n0 = nt * NTILE;

        // build z_cat tile (92 x 16)
        for (int idx = t; idx < KC * NTILE; idx += 256) {
            int k = idx >> 4, n = idx & 15;
            int ng = n0 + n;
            int i = ng / NJ;
            int j = ng - i * NJ;
            float val;
            if (j < NSEL) {
                if (k < DD) val = fabsf(z0s[k * NI + i] - z1hf[k * NSEL + j]);
                else { int c = k - DD; val = z0s[c * NI + i] * z1hf[c * NSEL + j]; }
            } else {
                int jj = j - NSEL;
                if (k < DD) val = fabsf(z0md[k * NI + i] - z1lf[k * NSEL + jj]);
                else { int c = k - DD; val = fabsf(z0mm[c * NI + i] - z1lf[c * NSEL + jj]); }
            }
            zc[idx] = val;
        }
        __syncthreads();

        // GEMM1: each wave does 2 M-tiles of 16 rows, K = 92 in 23 steps of 4
#pragma unroll
        for (int subm = 0; subm < 2; ++subm) {
            int m0 = (wv * 2 + subm) * 16;
            v8f acc = {0.f, 0.f, 0.f, 0.f, 0.f, 0.f, 0.f, 0.f};
            for (int ks = 0; ks < 23; ++ks) {
                int k0 = ks * 4;
                const float* wrow = conv_w + (m0 + l16) * KC + k0 + 2 * h;
                v2f a; a.x = wrow[0]; a.y = wrow[1];
                v2f b; b.x = zc[(k0 + 2 * h) * NTILE + l16];
                       b.y = zc[(k0 + 2 * h + 1) * NTILE + l16];
                acc = wmma4(a, b, acc);
            }
#pragma unroll
            for (int v = 0; v < 8; ++v) {
                int ch = m0 + v + 8 * h;
                float val = fmaxf(acc[v] + conv_b[ch], 0.f);
                if (FULL) {
                    c1t[ch * NTILE + l16] = val;
                } else {
                    sA[subm][v] += val;
                    qA[subm][v] += val * val;
                }
            }
        }
        __syncthreads();

        if (FULL) {
            // GEMM2: each wave does one M-tile of 16 rows (M=128), K = 256 in 64 steps
            int m0 = wv * 16;
            v8f acc = {0.f, 0.f, 0.f, 0.f, 0.f, 0.f, 0.f, 0.f};
            for (int ks = 0; ks < 64; ++ks) {
                int k0 = ks * 4;
                const float* wrow = w2f + (m0 + l16) * H1 + k0 + 2 * h;
                v2f a; a.x = wrow[0]; a.y = wrow[1];
                v2f b; b.x = c1t[(k0 + 2 * h) * NTILE + l16];
                       b.y = c1t[(k0 + 2 * h + 1) * NTILE + l16];
                acc = wmma4(a, b, acc);
            }
#pragma unroll
            for (int v = 0; v < 8; ++v) {
                int ch = m0 + v + 8 * h;
                float val = fmaxf(acc[v] + b2f[ch], 0.f);
                out[(long)ch * NPIX + n0 + l16] = val;
                s2[v] += val;
                q2[v] += val * val;
            }
            __syncthreads();
        }
    }

    // write per-block BN partials (no atomics): reduce over the 16 lanes of each half
    if (!FULL) {
        float* p1s = ws + OF_P1S;
        float* p1q = ws + OF_P1Q;
#pragma unroll
        for (int subm = 0; subm < 2; ++subm) {
#pragma unroll
            for (int v = 0; v < 8; ++v) {
                int ch = (wv * 2 + subm) * 16 + v + 8 * h;
                float s = sA[subm][v], q = qA[subm][v];
                for (int off = 1; off < 16; off <<= 1) {
                    s += __shfl_xor(s, off, 16);
                    q += __shfl_xor(q, off, 16);
                }
                if (l16 == 0) {
                    p1s[blockIdx.x * H1 + ch] = s;
                    p1q[blockIdx.x * H1 + ch] = q;
                }
            }
        }
    } else {
        float* p2s = ws + OF_P2S;
        float* p2q = ws + OF_P2Q;
#pragma unroll
        for (int v = 0; v < 8; ++v) {
            int ch = wv * 16 + v + 8 * h;
            float s = s2[v], q = q2[v];
            for (int off = 1; off < 16; off <<= 1) {
                s += __shfl_xor(s, off, 16);
                q += __shfl_xor(q, off, 16);
            }
            if (l16 == 0) {
                p2s[blockIdx.x * H2C + ch] = s;
                p2q[blockIdx.x * H2C + ch] = q;
            }
        }
    }
}

// ---------------- K4: finalize BN1 -> a1,d1 ----------------
__global__ __launch_bounds__(256)
void k_bn1(const float* __restrict__ g, const float* __restrict__ b, float* __restrict__ ws)
{
    int ch = threadIdx.x;
    float s = 0.f, q = 0.f;
    for (int blk = 0; blk < NBLK; ++blk) {
        s += ws[OF_P1S + blk * H1 + ch];
        q += ws[OF_P1Q + blk * H1 + ch];
    }
    float mean = s / (float)NPIX;
    float var  = q / (float)NPIX - mean * mean;
    float a = g[ch] * rsqrtf(var + 1e-5f);
    ws[OF_A1 + ch] = a;
    ws[OF_D1 + ch] = b[ch] - mean * a;
}

// ---------------- K4b: fold BN1 into conv2 weights ----------------
__global__ __launch_bounds__(256)
void k_fold(const float* __restrict__ conv2_w, const float* __restrict__ conv2_b, float* __restrict__ ws)
{
    __shared__ float red[H1];
    int o = blockIdx.x, c = threadIdx.x;
    float w  = conv2_w[o * H1 + c];
    float a1 = ws[OF_A1 + c], d1 = ws[OF_D1 + c];
    ws[OF_W2F + o * H1 + c] = w * a1;
    red[c] = w * d1;
    __syncthreads();
    for (int s = H1 / 2; s > 0; s >>= 1) {
        if (c < s) red[c] += red[c + s];
        __syncthreads();
    }
    if (c == 0) ws[OF_B2F + o] = conv2_b[o] + red[0];
}

// ---------------- K6: finalize BN2 -> a2,d2 ----------------
__global__ __launch_bounds__(128)
void k_bn2(const float* __restrict__ g, const float* __restrict__ b, float* __restrict__ ws)
{
    int ch = threadIdx.x;
    float s = 0.f, q = 0.f;
    for (int blk = 0; blk < NBLK; ++blk) {
        s += ws[OF_P2S + blk * H2C + ch];
        q += ws[OF_P2Q + blk * H2C + ch];
    }
    float mean = s / (float)NPIX;
    float var  = q / (float)NPIX - mean * mean;
    float a = g[ch] * rsqrtf(var + 1e-5f);
    ws[OF_A2 + ch] = a;
    ws[OF_D2 + ch] = b[ch] - mean * a;
}

// ---------------- K7: apply BN2 affine in-place ----------------
__global__ __launch_bounds__(256)
void k_norm(float* __restrict__ out, const float* __restrict__ ws)
{
    int ch = blockIdx.y;
    long pos = (long)blockIdx.x * blockDim.x + threadIdx.x;
    if (pos < NPIX) {
        float a = ws[OF_A2 + ch], d = ws[OF_D2 + ch];
        long idx = (long)ch * NPIX + pos;
        out[idx] = a * out[idx] + d;
    }
}

extern "C" void kernel_launch(void* const* d_in, const int* in_sizes, int n_in,
                              void* d_out, int out_size, void* d_ws, size_t ws_size,
                              hipStream_t stream)
{
    const float* z0      = (const float*)d_in[0];
    const float* z1      = (const float*)d_in[1];
    const float* we1     = (const float*)d_in[2];
    const float* we2     = (const float*)d_in[3];
    const float* we3     = (const float*)d_in[4];
    const float* we4     = (const float*)d_in[5];
    const float* we5     = (const float*)d_in[6];
    const float* conv_w  = (const float*)d_in[7];
    const float* conv_b  = (const float*)d_in[8];
    const float* bn_g    = (const float*)d_in[9];
    const float* bn_b    = (const float*)d_in[10];
    const float* conv2_w = (const float*)d_in[11];
    const float* conv2_b = (const float*)d_in[12];
    const float* bn2_g   = (const float*)d_in[13];
    const float* bn2_b   = (const float*)d_in[14];
    const int* cdrH      = (const int*)d_in[15];
    const int* cdrL      = (const int*)d_in[16];
    const int* notH      = (const int*)d_in[17];
    const int* notL      = (const int*)d_in[18];

    float* ws  = (float*)d_ws;
    float* out = (float*)d_out;

    k_pre1<<<1, 512, 0, stream>>>(z0, z1, we1, we2, we3, we4, we5,
                                  cdrH, cdrL, notH, notL,
                                  in_sizes[15], in_sizes[17], in_sizes[16], in_sizes[18], ws);
    k_pre2<<<NI, 64, 0, stream>>>(z0, ws);
    k_main<false><<<NBLK, 256, 0, stream>>>(conv_w, conv_b, ws, out);
    k_bn1<<<1, H1, 0, stream>>>(bn_g, bn_b, ws);
    k_fold<<<H2C, H1, 0, stream>>>(conv2_w, conv2_b, ws);
    k_main<true><<<NBLK, 256, 0, stream>>>(conv_w, conv_b, ws, out);
    k_bn2<<<1, H2C, 0, stream>>>(bn2_g, bn2_b, ws);
    k_norm<<<dim3((NPIX + 255) / 256, H2C), 256, 0, stream>>>(out, ws);
}